// E2PNBackbone_10196252360859
// MI455X (gfx1250) — compile-verified
//
#include <hip/hip_runtime.h>
#include <hip/hip_bf16.h>

typedef __attribute__((ext_vector_type(2))) float v2f_t;
typedef __attribute__((ext_vector_type(8))) float v8f_t;

#define NEG_SLOPE_F 0.1f
#define BIGF 3.0e38f

#if __has_builtin(__builtin_amdgcn_wmma_f32_16x16x4_f32)
#define HAVE_WMMA_F32X4 1
#else
#define HAVE_WMMA_F32X4 0
#endif

static inline int padded64(int n) { return (n + 63) & ~63; }

// ---------------------------------------------------------------- utilities
__global__ void fill_ones_kernel(float* __restrict__ p, int n) {
  int i = blockIdx.x * blockDim.x + threadIdx.x;
  if (i < n) p[i] = 1.0f;
}

__global__ void fill_zero_kernel(float* __restrict__ p, int n) {
  int i = blockIdx.x * blockDim.x + threadIdx.x;
  if (i < n) p[i] = 0.0f;
}

__global__ void copy_kernel(const float* __restrict__ s, float* __restrict__ d, int n) {
  int i = blockIdx.x * blockDim.x + threadIdx.x;
  if (i < n) d[i] = s[i];
}

// dst[m*K + k] = src[k*M + m]  (K x M  ->  M x K)
__global__ void transpose_kernel(const float* __restrict__ src, float* __restrict__ dst,
                                 int K, int M) {
  int i = blockIdx.x * blockDim.x + threadIdx.x;
  if (i < K * M) {
    int k = i / M;
    int m = i - k * M;
    dst[(size_t)m * K + k] = src[i];
  }
}

__global__ void subsample_kernel(const float* __restrict__ pin, float* __restrict__ pout,
                                 int nout, int stride) {
  int i = blockIdx.x * blockDim.x + threadIdx.x;
  if (i < nout) {
    pout[3 * i + 0] = pin[3 * i * stride + 0];
    pout[3 * i + 1] = pin[3 * i * stride + 1];
    pout[3 * i + 2] = pin[3 * i * stride + 2];
  }
}

// ------------------------------------------------------------------- KNN
// One 256-thread block per query. Distances to all candidates staged in LDS
// (max 8192 cands = 32KB; WGP has 320KB). k rounds of argmin extraction;
// ties resolve to the lower candidate index (matches jax top_k order).
__global__ void __launch_bounds__(256) knn_kernel(
    const float* __restrict__ q, const float* __restrict__ p,
    int ncand, int k, int* __restrict__ oidx, float* __restrict__ od2) {
  __shared__ float sd[8192];
  __shared__ float rv[256];
  __shared__ int ri[256];
  const int qi = blockIdx.x;
  const int tid = threadIdx.x;
  const float qx = q[qi * 3 + 0];
  const float qy = q[qi * 3 + 1];
  const float qz = q[qi * 3 + 2];
  for (int c = tid; c < ncand; c += 256) {
    float dx = p[c * 3 + 0] - qx;
    float dy = p[c * 3 + 1] - qy;
    float dz = p[c * 3 + 2] - qz;
    sd[c] = dx * dx + dy * dy + dz * dz;
  }
  __syncthreads();
  for (int r = 0; r < k; ++r) {
    float bv = BIGF;
    int bi = 0x7fffffff;
    for (int c = tid; c < ncand; c += 256) {
      float v = sd[c];
      if (v < bv || (v == bv && c < bi)) { bv = v; bi = c; }
    }
    rv[tid] = bv;
    ri[tid] = bi;
    __syncthreads();
    for (int s = 128; s > 0; s >>= 1) {
      if (tid < s) {
        float ov = rv[tid + s];
        int oi = ri[tid + s];
        if (ov < rv[tid] || (ov == rv[tid] && oi < ri[tid])) { rv[tid] = ov; ri[tid] = oi; }
      }
      __syncthreads();
    }
    if (tid == 0) {
      int w = ri[0];
      if (w == 0x7fffffff) w = 0;
      oidx[qi * k + r] = w;
      od2[qi * k + r] = rv[0];
      sd[w] = BIGF;
    }
    __syncthreads();
  }
}

// ------------------------------------------------------------- aggregation
// agg[n,c] = sum_k w[n,k] * F[idx[n,k],c] / (sum_k w + 1e-6)
// Weights AND neighbor indices staged in LDS once per point.
__global__ void __launch_bounds__(256) aggregate_kernel(
    const float* __restrict__ F, const int* __restrict__ idx,
    const float* __restrict__ d2, int C, float r2, float inv2s2,
    float* __restrict__ out) {
  __shared__ float w[64];
  __shared__ int sidx[64];
  __shared__ float winv;
  const int n = blockIdx.x;
  const int tid = threadIdx.x;
  if (tid < 64) {
    float d = d2[n * 64 + tid];
    w[tid] = (d < r2) ? expf(-d * inv2s2) : 0.0f;
    sidx[tid] = idx[n * 64 + tid];
  }
  __syncthreads();
  if (tid == 0) {
    float s = 0.0f;
    for (int i = 0; i < 64; ++i) s += w[i];
    winv = 1.0f / (s + 1e-6f);
  }
  __syncthreads();
  for (int c = tid; c < C; c += 256) {
    float acc = 0.0f;
    for (int kk = 0; kk < 64; ++kk) {
      acc += w[kk] * F[(size_t)sidx[kk] * C + c];
    }
    out[n * C + c] = acc * winv;
  }
}

// ------------------------------------------------------------- WMMA GEMM
// D = act(A @ W + bias) with W pre-transposed (Wt: M x K row-major).
// A: Npad x K row-major fp32, Npad = ceil64(N), pad rows zeroed by host seq.
// K % 4 == 0, M % 32 == 0.  One wave computes a 64x32 output strip
// (4 row tiles x 2 col tiles).  Per K-step: 4 A b64 loads + 2 B b64 loads
// feed 8 V_WMMA_F32_16X16X4_F32.  No guards in the hot loop; EXEC all-ones.
// Layouts per ISA 7.12.2:
//   A 16x4 : lanes 0-15 -> M=lane, K=k0+{0,1}; lanes 16-31 -> K=k0+{2,3}
//   B 4x16 : vgpr0 = rows k0 / k0+2, vgpr1 = rows k0+1 / k0+3, N=lane&15
//   C/D    : vgpr v -> row v (lanes<16) or v+8, col = lane&15
__global__ void __launch_bounds__(32) gemm_wmma_kernel(
    const float* __restrict__ A, const float* __restrict__ Wt,
    const float* __restrict__ bias, float* __restrict__ D,
    int N, int Kd, int M, int act) {
  const int lane = threadIdx.x;
  const int tm = blockIdx.x * 64;
  const int tn = blockIdx.y * 32;
  const int half = lane >> 4;
  const int l16 = lane & 15;
  const int bcol0 = tn + l16;
  const int bcol1 = tn + 16 + l16;
  v8f_t c00 = {}, c10 = {}, c20 = {}, c30 = {};
  v8f_t c01 = {}, c11 = {}, c21 = {}, c31 = {};
  const float* Ar = A + (size_t)(tm + l16) * Kd;
  const float* Br0 = Wt + (size_t)bcol0 * Kd;
  const float* Br1 = Wt + (size_t)bcol1 * Kd;
#if HAVE_WMMA_F32X4
  for (int k0 = 0; k0 < Kd; k0 += 4) {
    const int k = k0 + half * 2;
    v2f_t b0 = *(const v2f_t*)(Br0 + k);
    v2f_t b1 = *(const v2f_t*)(Br1 + k);
    v2f_t a0 = *(const v2f_t*)(Ar + k);
    v2f_t a1 = *(const v2f_t*)(Ar + (size_t)16 * Kd + k);
    v2f_t a2 = *(const v2f_t*)(Ar + (size_t)32 * Kd + k);
    v2f_t a3 = *(const v2f_t*)(Ar + (size_t)48 * Kd + k);
    c00 = __builtin_amdgcn_wmma_f32_16x16x4_f32(false, a0, false, b0, (short)0, c00, false, false);
    c10 = __builtin_amdgcn_wmma_f32_16x16x4_f32(false, a1, false, b0, (short)0, c10, false, false);
    c20 = __builtin_amdgcn_wmma_f32_16x16x4_f32(false, a2, false, b0, (short)0, c20, false, false);
    c30 = __builtin_amdgcn_wmma_f32_16x16x4_f32(false, a3, false, b0, (short)0, c30, false, false);
    c01 = __builtin_amdgcn_wmma_f32_16x16x4_f32(false, a0, false, b1, (short)0, c01, false, false);
    c11 = __builtin_amdgcn_wmma_f32_16x16x4_f32(false, a1, false, b1, (short)0, c11, false, false);
    c21 = __builtin_amdgcn_wmma_f32_16x16x4_f32(false, a2, false, b1, (short)0, c21, false, false);
    c31 = __builtin_amdgcn_wmma_f32_16x16x4_f32(false, a3, false, b1, (short)0, c31, false, false);
  }
#else
  for (int k = 0; k < Kd; ++k) {
    float bk0 = Wt[(size_t)bcol0 * Kd + k];
    float bk1 = Wt[(size_t)bcol1 * Kd + k];
    for (int v = 0; v < 8; ++v) {
      int r = v + half * 8;
      float a0 = A[(size_t)(tm + r) * Kd + k];
      float a1 = A[(size_t)(tm + 16 + r) * Kd + k];
      float a2 = A[(size_t)(tm + 32 + r) * Kd + k];
      float a3 = A[(size_t)(tm + 48 + r) * Kd + k];
      c00[v] += a0 * bk0; c10[v] += a1 * bk0; c20[v] += a2 * bk0; c30[v] += a3 * bk0;
      c01[v] += a0 * bk1; c11[v] += a1 * bk1; c21[v] += a2 * bk1; c31[v] += a3 * bk1;
    }
  }
#endif
  const float bb0 = bias ? bias[bcol0] : 0.0f;
  const float bb1 = bias ? bias[bcol1] : 0.0f;
  v8f_t acc0[4] = {c00, c10, c20, c30};
  v8f_t acc1[4] = {c01, c11, c21, c31};
  for (int t = 0; t < 4; ++t) {
    for (int v = 0; v < 8; ++v) {
      int row = tm + t * 16 + v + half * 8;
      if (row < N) {
        float x0 = acc0[t][v] + bb0;
        float x1 = acc1[t][v] + bb1;
        if (act) {
          x0 = (x0 > 0.0f) ? x0 : NEG_SLOPE_F * x0;
          x1 = (x1 > 0.0f) ? x1 : NEG_SLOPE_F * x1;
        }
        D[(size_t)row * M + bcol0] = x0;
        D[(size_t)row * M + bcol1] = x1;
      }
    }
  }
}

// ------------------------------------------------- layer0 (K=1) special case
// out[n,c] = lrelu(agg[n] * W[c] + b[c])
__global__ void layer0_linear_kernel(const float* __restrict__ agg,
                                     const float* __restrict__ W,
                                     const float* __restrict__ b,
                                     float* __restrict__ out, int N, int M) {
  int i = blockIdx.x * blockDim.x + threadIdx.x;
  if (i < N * M) {
    int row = i / M, col = i - row * M;
    float xv = agg[row] * W[col] + b[col];
    out[i] = (xv > 0.0f) ? xv : NEG_SLOPE_F * xv;
  }
}

// ------------------------------------------------------------- group norm
__global__ void __launch_bounds__(512) groupnorm_lrelu_kernel(
    float* __restrict__ h, const float* __restrict__ gamma,
    const float* __restrict__ beta, int C, int groups) {
  __shared__ float sh[512];
  const int n = blockIdx.x;
  const int c = threadIdx.x;
  const int gs = C / groups;
  float v = h[(size_t)n * C + c];
  sh[c] = v;
  __syncthreads();
  const int g0 = (c / gs) * gs;
  float mu = 0.0f, m2 = 0.0f;
  for (int i = 0; i < gs; ++i) {
    float xg = sh[g0 + i];
    mu += xg;
    m2 += xg * xg;
  }
  mu /= gs;
  m2 /= gs;
  float var = m2 - mu * mu;
  float hn = (v - mu) * rsqrtf(var + 1e-5f);
  float y = hn * gamma[c] + beta[c];
  h[(size_t)n * C + c] = (y > 0.0f) ? y : NEG_SLOPE_F * y;
}

// ------------------------------------------------------------- concat
__global__ void concat_gather_kernel(const float* __restrict__ a,
                                     const int* __restrict__ gidx, int Ca,
                                     const float* __restrict__ b, int Cb,
                                     float* __restrict__ out, int n) {
  int i = blockIdx.x * blockDim.x + threadIdx.x;
  int Ct = Ca + Cb;
  if (i < n * Ct) {
    int r = i / Ct;
    int col = i - r * Ct;
    out[i] = (col < Ca) ? a[(size_t)gidx[r] * Ca + col]
                        : b[(size_t)r * Cb + (col - Ca)];
  }
}

// =========================================================== host driver
extern "C" void kernel_launch(void* const* d_in, const int* in_sizes, int n_in,
                              void* d_out, int out_size, void* d_ws, size_t ws_size,
                              hipStream_t stream) {
  const float* x = (const float*)d_in[0];
  const float* Wl[11];
  const float* Bl[11];
  for (int i = 0; i < 11; ++i) {
    Wl[i] = (const float*)d_in[1 + 2 * i];
    Bl[i] = (const float*)d_in[2 + 2 * i];
  }
  const float* iW[4];
  const float* iB[4];
  for (int j = 0; j < 4; ++j) {
    iW[j] = (const float*)d_in[23 + 2 * j];
    iB[j] = (const float*)d_in[24 + 2 * j];
  }
  const float* dec3W = (const float*)d_in[31];
  const float* dec3b = (const float*)d_in[32];
  const float* gnG = (const float*)d_in[33];
  const float* gnB = (const float*)d_in[34];
  const float* dec2W = (const float*)d_in[35];
  const float* dec2b = (const float*)d_in[36];

  const int MLPS[11] = {64, 128, 128, 256, 256, 256, 512, 512, 512, 1024, 1024};
  const int STRIDES[11] = {1, 1, 3, 1, 1, 3, 1, 1, 3, 1, 1};
  const int MULT[11] = {1, 1, 1, 2, 2, 2, 4, 4, 4, 8, 8};
  const int N0 = 8192, N1 = 2730, N2 = 910, N3 = 303;

  int Ns[11];
  {
    int n = N0;
    for (int i = 0; i < 11; ++i) { n /= STRIDES[i]; Ns[i] = n; }
  }

  // ------------------------ workspace layout (floats, 16B aligned slices)
  float* wsf = (float*)d_ws;
  size_t off = 0;
  auto walloc = [&](size_t n) {
    float* p = wsf + off;
    off += (n + 3) & ~(size_t)3;
    return p;
  };

  // Feature buffers padded to 64 rows (store layers feed WMMA GEMMs as A).
  float* F[11];
  for (int i = 0; i < 11; ++i) F[i] = walloc((size_t)padded64(Ns[i]) * MLPS[i]);
  float* ONES = walloc(N0);
  float* P1 = walloc((size_t)N1 * 3);
  float* P2 = walloc((size_t)N2 * 3);
  float* P3 = walloc((size_t)N3 * 3);
  int* KIDX = (int*)walloc((size_t)N0 * 64);
  float* KD2 = walloc((size_t)N0 * 64);
  float* AGG = walloc((size_t)1 << 20);  // covers max padded64(N)*Cin = 704512
  float* INV0 = walloc((size_t)N0 * 128);
  float* INV1 = walloc((size_t)N1 * 256);
  float* INV2 = walloc((size_t)N2 * 512);
  float* INV3 = walloc((size_t)N3 * 1024);
  float* LAT3IN = walloc((size_t)padded64(N2) * 1536);
  float* LAT3 = walloc((size_t)N2 * 512);
  float* LAT2IN = walloc((size_t)padded64(N1) * 768);
  int* IDX3 = (int*)walloc(N2);
  int* IDX2 = (int*)walloc(N1);
  // Transposed weights (M x K) for contiguous b64 B-fragment loads.
  float* WlT[11];
  WlT[0] = nullptr;  // layer0 uses the rank-1 kernel
  for (int i = 1; i < 11; ++i) WlT[i] = walloc((size_t)MLPS[i - 1] * MLPS[i]);
  float* iWT[4];
  const int storeL[4] = {1, 4, 7, 10};
  for (int j = 0; j < 4; ++j) iWT[j] = walloc((size_t)MLPS[storeL[j]] * MLPS[storeL[j]]);
  float* dec3WT = walloc((size_t)1536 * 512);
  float* dec2WT = walloc((size_t)768 * 256);
  if (off * sizeof(float) > ws_size) return;  // workspace too small

  auto zero_pad = [&](float* base, int nrows, int C) {
    int pad = (padded64(nrows) - nrows) * C;
    if (pad > 0)
      fill_zero_kernel<<<(pad + 255) / 256, 256, 0, stream>>>(base + (size_t)nrows * C, pad);
  };
  auto do_transpose = [&](const float* src, float* dst, int K, int M) {
    int tot = K * M;
    transpose_kernel<<<(tot + 255) / 256, 256, 0, stream>>>(src, dst, K, M);
  };

  // ------------------------ d_out layout (floats)
  float* out = (float*)d_out;
  const size_t O_FEATC = 0;                // 303*1024
  const size_t O_FEATF = 310272;           // 2730*256
  const size_t O_PTSC = O_FEATF + 698880;  // 303*3
  const size_t O_PTSF = O_PTSC + 909;      // 2730*3
  const size_t O_FINV = O_PTSF + 8190;     // 303*1024

  // ------------------------ weight transposes (small; hit L2 thereafter)
  for (int i = 1; i < 11; ++i) do_transpose(Wl[i], WlT[i], MLPS[i - 1], MLPS[i]);
  for (int j = 0; j < 4; ++j) do_transpose(iW[j], iWT[j], MLPS[storeL[j]], MLPS[storeL[j]]);
  do_transpose(dec3W, dec3WT, 1536, 512);
  do_transpose(dec2W, dec2WT, 768, 256);

  // ------------------------ encoder
  fill_ones_kernel<<<(N0 + 255) / 256, 256, 0, stream>>>(ONES, N0);

  const float* Pcur = x;
  int ncur = N0;
  for (int i = 0; i < 11; ++i) {
    const int nout = Ns[i];
    const int Cin = (i == 0) ? 1 : MLPS[i - 1];
    const int Cout = MLPS[i];
    const float radius = 2.5f * 0.025f * MULT[i];
    const float sigma = 2.0f * 0.025f * MULT[i];
    const float r2 = radius * radius;
    const float inv2s2 = 1.0f / (2.0f * sigma * sigma);

    const float* Pout = Pcur;
    if (STRIDES[i] > 1) {
      float* dst = (i == 2) ? P1 : (i == 5) ? P2 : P3;
      subsample_kernel<<<(nout + 255) / 256, 256, 0, stream>>>(Pcur, dst, nout, STRIDES[i]);
      Pout = dst;
    }
    knn_kernel<<<nout, 256, 0, stream>>>(Pout, Pcur, ncur, 64, KIDX, KD2);
    const float* Fin = (i == 0) ? ONES : F[i - 1];
    aggregate_kernel<<<nout, 256, 0, stream>>>(Fin, KIDX, KD2, Cin, r2, inv2s2, AGG);
    if (i == 0) {
      layer0_linear_kernel<<<(nout * Cout + 255) / 256, 256, 0, stream>>>(
          AGG, Wl[0], Bl[0], F[0], nout, Cout);
    } else {
      zero_pad(AGG, nout, Cin);  // AGG is reused across layers; re-zero pad rows
      dim3 gg(padded64(nout) / 64, Cout / 32);
      gemm_wmma_kernel<<<gg, 32, 0, stream>>>(AGG, WlT[i], Bl[i], F[i], nout, Cin, Cout, 1);
    }
    Pcur = Pout;
    ncur = nout;
  }

  // ------------------------ invariant linears on stored features
  float* INVp[4] = {INV0, INV1, INV2, INV3};
  for (int j = 0; j < 4; ++j) {
    int li = storeL[j];
    int Nst = Ns[li];
    int C = MLPS[li];
    zero_pad(F[li], Nst, C);
    dim3 gg(padded64(Nst) / 64, C / 32);
    gemm_wmma_kernel<<<gg, 32, 0, stream>>>(F[li], iWT[j], iB[j], INVp[j], Nst, C, C, 0);
  }

  // ------------------------ decoder
  knn_kernel<<<N2, 256, 0, stream>>>(P2, P3, N3, 1, IDX3, KD2);
  {
    int tot = N2 * 1536;
    concat_gather_kernel<<<(tot + 255) / 256, 256, 0, stream>>>(INV3, IDX3, 1024, INV2, 512, LAT3IN, N2);
    zero_pad(LAT3IN, N2, 1536);
    dim3 gg(padded64(N2) / 64, 512 / 32);
    gemm_wmma_kernel<<<gg, 32, 0, stream>>>(LAT3IN, dec3WT, dec3b, LAT3, N2, 1536, 512, 0);
  }
  groupnorm_lrelu_kernel<<<N2, 512, 0, stream>>>(LAT3, gnG, gnB, 512, 32);

  knn_kernel<<<N1, 256, 0, stream>>>(P1, P2, N2, 1, IDX2, KD2);
  {
    int tot = N1 * 768;
    concat_gather_kernel<<<(tot + 255) / 256, 256, 0, stream>>>(LAT3, IDX2, 512, INV1, 256, LAT2IN, N1);
    zero_pad(LAT2IN, N1, 768);
    dim3 gg(padded64(N1) / 64, 256 / 32);
    gemm_wmma_kernel<<<gg, 32, 0, stream>>>(LAT2IN, dec2WT, dec2b, out + O_FEATF, N1, 768, 256, 0);
  }

  // ------------------------ remaining outputs
  copy_kernel<<<(303 * 1024 + 255) / 256, 256, 0, stream>>>(F[10], out + O_FEATC, 303 * 1024);
  copy_kernel<<<(909 + 255) / 256, 256, 0, stream>>>(P3, out + O_PTSC, 909);
  copy_kernel<<<(8190 + 255) / 256, 256, 0, stream>>>(P1, out + O_PTSF, 8190);
  copy_kernel<<<(303 * 1024 + 255) / 256, 256, 0, stream>>>(INV3, out + O_FINV, 303 * 1024);
}